// KalmanSmoother_71794673319974
// MI455X (gfx1250) — compile-verified
//
#include <hip/hip_runtime.h>
#include <hip/hip_bf16.h>
#include <stdint.h>

// ---------------------------------------------------------------------------
// Per-column linear interpolation of masked gaps (Kalman smoother interp).
//   X    : [T, D] float32 row-major
//   mask : [T, D] int (nonzero = observed)
//   out  : [T, D] float32
// Decoupled 3-phase chunked scan along T (chunk = 32 rows):
//   Phase A: per (chunk, 4 cols) -> packed 32-bit mask words (b128 loads)
//            + first/last observed (idx, val) per column in the chunk.
//   Phase B: per col -> cross-chunk prefix (prev observed) / suffix (next
//            observed) boundary state, column first-value / has-observation.
//   Phase C: per WG = 32x128 f32 tile staged to LDS via
//            GLOBAL_LOAD_ASYNC_TO_LDS_B128 (ASYNCcnt), register backward scan
//            for next-observed, forward scan writes results into the tile,
//            tile streamed out via GLOBAL_STORE_ASYNC_FROM_LDS_B128.
// HBM traffic ~ X read once + mask read once + out written once (~390 MB)
// => ~17 us at 23.3 TB/s. No matrix math exists here, so WMMA is inapplicable;
// the CDNA5-specific leverage is wave32 + async LDS DMA + prefetch.
// ---------------------------------------------------------------------------

namespace {
constexpr int Tn = 16384;
constexpr int Dn = 2048;           // 2^11
constexpr int C  = 32;             // chunk length == bits in one mask word
constexpr int NC = Tn / C;         // 512 chunks per column
constexpr int NR = NC * Dn;        // 1,048,576 (chunk, col) records
constexpr int TW = 128;            // Phase-C tile width (columns per WG)
static_assert((Dn & (Dn - 1)) == 0, "Dn must be a power of two");
}

#define AS1 __attribute__((address_space(1)))
#define AS3 __attribute__((address_space(3)))

// Exact parameter type of the async-LDS builtins per hipcc diagnostic:
// pointer to GCC-vector "int __vector_size__(16)" in AS1 (global) / AS3 (LDS).
typedef int v4i __attribute__((__vector_size__(16)));

#if __has_builtin(__builtin_amdgcn_global_load_async_to_lds_b128)
#define HAVE_ASYNC_LOAD 1
#else
#define HAVE_ASYNC_LOAD 0
#endif
#if __has_builtin(__builtin_amdgcn_global_store_async_from_lds_b128)
#define HAVE_ASYNC_STORE 1
#else
#define HAVE_ASYNC_STORE 0
#endif

__device__ __forceinline__ void wait_asynccnt0() {
#if __has_builtin(__builtin_amdgcn_s_wait_asynccnt)
    __builtin_amdgcn_s_wait_asynccnt(0);
#else
    asm volatile("s_wait_asynccnt 0" ::: "memory");
#endif
}

// ------------------------- Phase A: chunk summaries -------------------------
// One thread handles one chunk x 4 adjacent columns: mask read as b128.
__global__ void ks_chunk_summary(const float* __restrict__ X,
                                 const int* __restrict__ mask,
                                 int* __restrict__ ckFirstI, float* __restrict__ ckFirstV,
                                 int* __restrict__ ckLastI,  float* __restrict__ ckLastV,
                                 unsigned* __restrict__ mbits) {
    int tid = blockIdx.x * blockDim.x + threadIdx.x;     // NR/4 threads
    int dg  = tid & (Dn / 4 - 1);                        // 512 column-quads
    int c   = tid >> 9;                                  // log2(Dn/4)
    int d0  = dg * 4;
    int base = c * C;
    const int* mp = mask + base * Dn + d0;

    unsigned b0 = 0, b1 = 0, b2 = 0, b3 = 0;
#pragma unroll
    for (int i = 0; i < C; ++i) {
        if ((i & 7) == 0)
            __builtin_prefetch(mp + (i + 16) * Dn, 0, 1);    // global_prefetch_b8
        int4 m = *(const int4*)(mp + i * Dn);                // b128 coalesced
        if (m.x) b0 |= (1u << i);
        if (m.y) b1 |= (1u << i);
        if (m.z) b2 |= (1u << i);
        if (m.w) b3 |= (1u << i);
    }

    int r = c * Dn + d0;
    *(uint4*)&mbits[r] = make_uint4(b0, b1, b2, b3);

    unsigned bs[4] = { b0, b1, b2, b3 };
    int   fI[4], lI[4];
    float fV[4], lV[4];
#pragma unroll
    for (int j = 0; j < 4; ++j) {
        if (bs[j]) {
            fI[j] = base + __builtin_ctz(bs[j]);
            lI[j] = base + 31 - __builtin_clz(bs[j]);
            fV[j] = X[fI[j] * Dn + d0 + j];   // scattered, L2-resident for Phase C
            lV[j] = X[lI[j] * Dn + d0 + j];
        } else {
            fI[j] = -1; lI[j] = -1; fV[j] = 0.f; lV[j] = 0.f;
        }
    }
    *(int4*)  &ckFirstI[r] = make_int4(fI[0], fI[1], fI[2], fI[3]);
    *(float4*)&ckFirstV[r] = make_float4(fV[0], fV[1], fV[2], fV[3]);
    *(int4*)  &ckLastI[r]  = make_int4(lI[0], lI[1], lI[2], lI[3]);
    *(float4*)&ckLastV[r]  = make_float4(lV[0], lV[1], lV[2], lV[3]);
}

// ----------------- Phase B: per-column cross-chunk boundary scan ------------
__global__ void ks_col_scan(const int* __restrict__ ckFirstI, const float* __restrict__ ckFirstV,
                            const int* __restrict__ ckLastI,  const float* __restrict__ ckLastV,
                            int* __restrict__ pbPrevI, float* __restrict__ pbPrevV,
                            int* __restrict__ pbNextI, float* __restrict__ pbNextV,
                            float* __restrict__ colFirstV, int* __restrict__ colHasV) {
    int d = blockIdx.x * blockDim.x + threadIdx.x;
    if (d >= Dn) return;

    int pI = -1; float pV = 0.f;
    for (int c = 0; c < NC; ++c) {
        int r = c * Dn + d;
        pbPrevI[r] = pI; pbPrevV[r] = pV;
        int li = ckLastI[r];
        if (li >= 0) { pI = li; pV = ckLastV[r]; }
    }
    int nI = Tn; float nV = 0.f;
    for (int c = NC - 1; c >= 0; --c) {
        int r = c * Dn + d;
        pbNextI[r] = nI; pbNextV[r] = nV;
        int fi = ckFirstI[r];
        if (fi >= 0) { nI = fi; nV = ckFirstV[r]; }
    }
    colFirstV[d] = nV;
    colHasV[d]   = (nI < Tn) ? 1 : 0;
}

// --------------------------- Phase C: fill output ---------------------------
__global__ void __launch_bounds__(TW) ks_fill(
        const float* __restrict__ X, float* __restrict__ out,
        const unsigned* __restrict__ mbits,
        const int* __restrict__ pbPrevI, const float* __restrict__ pbPrevV,
        const int* __restrict__ pbNextI, const float* __restrict__ pbNextV,
        const float* __restrict__ colFirstV, const int* __restrict__ colHasV) {
    __shared__ float tile[C * TW];                 // 16 KB: 32 rows x 128 cols

    int b    = blockIdx.x;
    int dg   = b & (Dn / TW - 1);                  // 16 column tiles
    int c    = b >> 4;                             // log2(Dn/TW)
    int d0   = dg * TW;
    int tc   = threadIdx.x;                        // 0..127 -> column in tile
    int d    = d0 + tc;
    int base = c * C;

    const float* gbase = X + base * Dn + d0;

    // ---- stage X tile into LDS (async DMA path on CDNA5) ----
#if HAVE_ASYNC_LOAD
#pragma unroll
    for (int k = 0; k < 8; ++k) {
        int idx = k * TW + tc;                     // float4 index 0..1023
        int rr  = idx >> 5;                        // tile row
        int ss  = idx & 31;                        // float4 within row
        __builtin_amdgcn_global_load_async_to_lds_b128(
            (AS1 v4i*)(gbase + rr * Dn + ss * 4),
            (AS3 v4i*)(&tile[idx * 4]), 0, 0);
    }
    wait_asynccnt0();
    __syncthreads();
#else
#pragma unroll
    for (int k = 0; k < 8; ++k) {
        int idx = k * TW + tc;
        int rr  = idx >> 5;
        int ss  = idx & 31;
        float4 v = *(const float4*)(gbase + rr * Dn + ss * 4);
        *(float4*)&tile[idx * 4] = v;
    }
    __syncthreads();
#endif

    int r = c * Dn + d;
    unsigned bits = mbits[r];

    // Backward register scan: next observed (idx,val) at index > t.
    int   nIa[C];
    float nVa[C];
    int   nI = pbNextI[r];
    float nV = pbNextV[r];
#pragma unroll
    for (int i = C - 1; i >= 0; --i) {
        nIa[i] = nI;
        nVa[i] = nV;
        if ((bits >> i) & 1u) { nI = base + i; nV = tile[i * TW + tc]; }
    }

    // Forward pass: carry prev observed, write result back into the tile.
    int   pI   = pbPrevI[r];
    float pV   = pbPrevV[r];
    float fV   = colFirstV[d];
    bool  hasV = colHasV[d] != 0;
#pragma unroll
    for (int i = 0; i < C; ++i) {
        float x = tile[i * TW + tc];
        bool  m = (bits >> i) & 1u;
        float o;
        if (m) {
            o = x;                                  // observed: keep value
        } else if (pI < 0) {
            o = fV;                                 // before first observation
        } else if (nIa[i] >= Tn) {
            o = pV;                                 // after last observation
        } else {
            float ratio = (float)(base + i - pI) / (float)(nIa[i] - pI);
            o = pV + ratio * (nVa[i] - pV);         // linear interpolation
        }
        if (!hasV) o = x;                           // column with no obs: copy
        tile[i * TW + tc] = o;                      // own column only: no hazard
        if (m) { pI = base + i; pV = x; }
    }
    __syncthreads();

    // ---- stream tile out (async LDS->global path on CDNA5) ----
    float* obase = out + base * Dn + d0;
#if HAVE_ASYNC_STORE
#pragma unroll
    for (int k = 0; k < 8; ++k) {
        int idx = k * TW + tc;
        int rr  = idx >> 5;
        int ss  = idx & 31;
        __builtin_amdgcn_global_store_async_from_lds_b128(
            (AS1 v4i*)(obase + rr * Dn + ss * 4),
            (AS3 v4i*)(&tile[idx * 4]), 0, 0);
    }
    // s_endpgm performs an implicit wait-idle; nothing reuses the tile.
#else
#pragma unroll
    for (int k = 0; k < 8; ++k) {
        int idx = k * TW + tc;
        int rr  = idx >> 5;
        int ss  = idx & 31;
        float4 v = *(const float4*)&tile[idx * 4];
        *(float4*)(obase + rr * Dn + ss * 4) = v;
    }
#endif
}

// ------------- Fallback (tiny workspace): one thread per column -------------
__global__ void ks_fallback(const float* __restrict__ X,
                            const int* __restrict__ mask,
                            float* __restrict__ out) {
    int d = blockIdx.x * blockDim.x + threadIdx.x;
    if (d >= Dn) return;
    int pI = -1; float pV = 0.f;
    int gapStart = 0; bool any = false;
    for (int t = 0; t < Tn; ++t) {
        __builtin_prefetch(X + (t + 8) * Dn + d, 0, 1);
        int   m = mask[t * Dn + d];
        float x = X[t * Dn + d];
        if (m != 0) {
            if (pI < 0) {
                for (int u = gapStart; u < t; ++u) out[u * Dn + d] = x;
            } else {
                float inv = 1.f / (float)(t - pI);
                for (int u = gapStart; u < t; ++u)
                    out[u * Dn + d] = pV + (float)(u - pI) * inv * (x - pV);
            }
            out[t * Dn + d] = x;
            pI = t; pV = x; gapStart = t + 1; any = true;
        }
    }
    if (!any) {
        for (int t = 0; t < Tn; ++t) out[t * Dn + d] = X[t * Dn + d];
    } else {
        for (int u = gapStart; u < Tn; ++u) out[u * Dn + d] = pV;
    }
}

// ---------------------------------------------------------------------------
extern "C" void kernel_launch(void* const* d_in, const int* in_sizes, int n_in,
                              void* d_out, int out_size, void* d_ws, size_t ws_size,
                              hipStream_t stream) {
    (void)in_sizes; (void)n_in; (void)out_size;
    const float* X = (const float*)d_in[0];
    const int*   M = (const int*)d_in[1];
    float* out     = (float*)d_out;

    char* ws = (char*)d_ws;
    const size_t rec  = (size_t)NR * sizeof(int);          // 4 MB per array
    const size_t need = rec * 9 + 2 * (size_t)Dn * sizeof(int);

    if (ws && ws_size >= need) {
        int*      ckFirstI  = (int*)     (ws + 0 * rec);
        float*    ckFirstV  = (float*)   (ws + 1 * rec);
        int*      ckLastI   = (int*)     (ws + 2 * rec);
        float*    ckLastV   = (float*)   (ws + 3 * rec);
        int*      pbPrevI   = (int*)     (ws + 4 * rec);
        float*    pbPrevV   = (float*)   (ws + 5 * rec);
        int*      pbNextI   = (int*)     (ws + 6 * rec);
        float*    pbNextV   = (float*)   (ws + 7 * rec);
        unsigned* mbits     = (unsigned*)(ws + 8 * rec);
        float*    colFirstV = (float*)   (ws + 9 * rec);
        int*      colHasV   = (int*)     (ws + 9 * rec + Dn * sizeof(float));

        hipLaunchKernelGGL(ks_chunk_summary, dim3((NR / 4) / 256), dim3(256), 0, stream,
                           X, M, ckFirstI, ckFirstV, ckLastI, ckLastV, mbits);
        hipLaunchKernelGGL(ks_col_scan, dim3(Dn / 256), dim3(256), 0, stream,
                           ckFirstI, ckFirstV, ckLastI, ckLastV,
                           pbPrevI, pbPrevV, pbNextI, pbNextV, colFirstV, colHasV);
        hipLaunchKernelGGL(ks_fill, dim3(NC * (Dn / TW)), dim3(TW), 0, stream,
                           X, out, mbits, pbPrevI, pbPrevV, pbNextI, pbNextV,
                           colFirstV, colHasV);
    } else {
        hipLaunchKernelGGL(ks_fallback, dim3(Dn / 256), dim3(256), 0, stream, X, M, out);
    }
}